// Mask_Feat_Loss_14980845929080
// MI455X (gfx1250) — compile-verified
//
#include <hip/hip_runtime.h>
#include <cstdint>
#include <cstddef>

#define B_      4
#define C_      256
#define H_      180
#define W_      180
#define NBOX    64
#define HW_     (H_ * W_)          // 32400
#define W4_     (W_ / 4)           // 45
#define NQUAD   (B_ * H_ * W4_)    // 32400
#define THREADS 256
#define NBLK    ((NQUAD + THREADS - 1) / THREADS)  // 127
#define CC      4                  // channel chunks (gridDim.y of stream pass)
#define CL      (C_ / CC)          // 64 channels per chunk

// Workspace layout (bytes)
#define WS_SSQ_BYTES  ((size_t)CC * NQUAD * 16)   // v4f per (chunk, quad)
#define WS_NZ_BYTES   ((size_t)CC * NQUAD * 4)    // nz nibble per (chunk, quad)
#define WS_PART_BYTES ((size_t)NBLK * 2 * 4)
#define WS_TOTAL      (WS_SSQ_BYTES + WS_NZ_BYTES + WS_PART_BYTES)

#ifndef __has_builtin
#define __has_builtin(x) 0
#endif

#if __has_builtin(__builtin_amdgcn_global_load_async_to_lds_b128)
#define HAS_ASYNC_B128 1
#else
#define HAS_ASYNC_B128 0
#endif

#if __has_builtin(__builtin_amdgcn_s_wait_asynccnt)
#define HAS_WAIT_ASYNC 1
#else
#define HAS_WAIT_ASYNC 0
#endif

// Native clang vector types (required by async-LDS and nontemporal builtins).
typedef int   v4i __attribute__((ext_vector_type(4)));
typedef float v4f __attribute__((ext_vector_type(4)));

#if HAS_ASYNC_B128
typedef v4i __attribute__((address_space(1))) gv4i;
typedef v4i __attribute__((address_space(3))) lv4i;
#endif

// ---------------------------------------------------------------------------
// Async global->LDS broadcast of the 4 KB box table (one b128 per thread),
// then wait asynccnt and barrier.
// ---------------------------------------------------------------------------
__device__ __forceinline__ void stage_boxes(const int* __restrict__ gt_boxes,
                                            int4* s_boxes, int tid)
{
#if HAS_ASYNC_B128
    {
        gv4i* gsrc = (gv4i*)(uintptr_t)(gt_boxes + (size_t)tid * 4);
        lv4i* ldst = (lv4i*)(uint32_t)(uintptr_t)(&s_boxes[tid]);
        __builtin_amdgcn_global_load_async_to_lds_b128(gsrc, ldst, 0, 0);
    }
#if HAS_WAIT_ASYNC
    __builtin_amdgcn_s_wait_asynccnt(0);
#else
    asm volatile("s_wait_asynccnt 0x0" ::: "memory");
#endif
#else
    s_boxes[tid] = ((const int4*)gt_boxes)[tid];
#endif
    __syncthreads();
}

// 4-bit coverage mask of pixels (h, w0..w0+3); x-interval is [rb_x, lt_x).
__device__ __forceinline__ unsigned quad_mask(const int4* __restrict__ bb,
                                              int h, int w0)
{
    unsigned qmask = 0u;
    for (int n = 0; n < NBOX; ++n) {
        const int4 bx = bb[n];              // {lt_x, lt_y, rb_x, rb_y}
        if (h >= bx.y && h < bx.w) {
#pragma unroll
            for (int j = 0; j < 4; ++j) {
                const int wj = w0 + j;
                qmask |= ((wj >= bx.z) && (wj < bx.x)) ? (1u << j) : 0u;
            }
            if (qmask == 0xFu) break;
        }
    }
    return qmask;
}

__device__ __forceinline__ void decode_quad(int qid, int& b, int& h, int& w0)
{
    b = qid / (H_ * W4_);
    const int r = qid - b * (H_ * W4_);
    h = r / W4_;
    w0 = (r - h * W4_) * 4;
}

// ---------------------------------------------------------------------------
// Stream pass: grid (NBLK, CC). Block (g,k) streams channels [k*CL,(k+1)*CL)
// for 256 quads; writes per-quad partial (ssq4, nz nibble). Skips unmasked
// quads (mask recomputed from LDS box table; combine recomputes identically).
// ---------------------------------------------------------------------------
extern "C" __global__ void __launch_bounds__(THREADS)
mfl_stream(const float* __restrict__ input, const float* __restrict__ target,
           const int* __restrict__ gt_boxes,
           v4f* __restrict__ ssq_out, unsigned* __restrict__ nz_out)
{
    __shared__ int4 s_boxes[B_ * NBOX];
    const int tid = threadIdx.x;
    stage_boxes(gt_boxes, s_boxes, tid);

    const int qid = blockIdx.x * THREADS + tid;
    if (qid >= NQUAD) return;

    int b, h, w0;
    decode_quad(qid, b, h, w0);
    const unsigned qmask = quad_mask(&s_boxes[b * NBOX], h, w0);
    if (!qmask) return;                     // combine skips these too

    const int chunk = blockIdx.y;
    const size_t base = (size_t)b * ((size_t)C_ * HW_)
                      + (size_t)(chunk * CL) * HW_
                      + (size_t)h * W_ + w0;
    const float* xp = input + base;
    const float* tp = target + base;

    float s0 = 0.f, s1 = 0.f, s2 = 0.f, s3 = 0.f;
    unsigned nz = 0u;
#pragma unroll 8
    for (int c = 0; c < CL; ++c) {
        const v4f xv = __builtin_nontemporal_load((const v4f*)(xp + (size_t)c * HW_));
        const v4f tv = __builtin_nontemporal_load((const v4f*)(tp + (size_t)c * HW_));
        // any(t != 0): NaN != 0 is true, matching the reference
        nz |= (tv.x != 0.f ? 1u : 0u) | (tv.y != 0.f ? 2u : 0u)
            | (tv.z != 0.f ? 4u : 0u) | (tv.w != 0.f ? 8u : 0u);
        // t = isnan(t) ? x : t  ==>  d = isnan(t) ? 0 : x - t
        const float d0 = (tv.x == tv.x) ? (xv.x - tv.x) : 0.f;
        const float d1 = (tv.y == tv.y) ? (xv.y - tv.y) : 0.f;
        const float d2 = (tv.z == tv.z) ? (xv.z - tv.z) : 0.f;
        const float d3 = (tv.w == tv.w) ? (xv.w - tv.w) : 0.f;
        s0 = fmaf(d0, d0, s0);
        s1 = fmaf(d1, d1, s1);
        s2 = fmaf(d2, d2, s2);
        s3 = fmaf(d3, d3, s3);
    }
    v4f ssq; ssq.x = s0; ssq.y = s1; ssq.z = s2; ssq.w = s3;
    ssq_out[(size_t)chunk * NQUAD + qid] = ssq;     // L2-resident intermediates
    nz_out[(size_t)chunk * NQUAD + qid] = nz;
}

// ---------------------------------------------------------------------------
// Combine pass: per quad, sum the CC chunk partials in fixed order, OR the nz
// bits, apply pos = mask & nz, block tree-reduce -> per-block (S, N).
// ---------------------------------------------------------------------------
extern "C" __global__ void __launch_bounds__(THREADS)
mfl_combine(const int* __restrict__ gt_boxes,
            const v4f* __restrict__ ssq_in, const unsigned* __restrict__ nz_in,
            float* __restrict__ partials)
{
    __shared__ int4  s_boxes[B_ * NBOX];
    __shared__ float sS[THREADS];
    __shared__ float sN[THREADS];

    const int tid = threadIdx.x;
    stage_boxes(gt_boxes, s_boxes, tid);

    const int qid = blockIdx.x * THREADS + tid;
    float tS = 0.0f, tN = 0.0f;

    if (qid < NQUAD) {
        int b, h, w0;
        decode_quad(qid, b, h, w0);
        const unsigned qmask = quad_mask(&s_boxes[b * NBOX], h, w0);
        if (qmask) {
            v4f ssq; ssq.x = 0.f; ssq.y = 0.f; ssq.z = 0.f; ssq.w = 0.f;
            unsigned nz = 0u;
#pragma unroll
            for (int k = 0; k < CC; ++k) {
                ssq += ssq_in[(size_t)k * NQUAD + qid];   // fixed order
                nz  |= nz_in[(size_t)k * NQUAD + qid];
            }
            const unsigned act = qmask & nz;
            if (act & 1u) { tS += ssq.x; tN += 1.0f; }
            if (act & 2u) { tS += ssq.y; tN += 1.0f; }
            if (act & 4u) { tS += ssq.z; tN += 1.0f; }
            if (act & 8u) { tS += ssq.w; tN += 1.0f; }
        }
    }

    sS[tid] = tS;
    sN[tid] = tN;
    __syncthreads();
    for (int s = THREADS / 2; s > 0; s >>= 1) {
        if (tid < s) {
            sS[tid] += sS[tid + s];
            sN[tid] += sN[tid + s];
        }
        __syncthreads();
    }
    if (tid == 0) {
        partials[2 * blockIdx.x + 0] = sS[0];
        partials[2 * blockIdx.x + 1] = sN[0];
    }
}

// ---------------------------------------------------------------------------
// Fallback single-pass kernel (used only if ws_size < WS_TOTAL).
// ---------------------------------------------------------------------------
extern "C" __global__ void __launch_bounds__(THREADS)
mfl_fused(const float* __restrict__ input, const float* __restrict__ target,
          const int* __restrict__ gt_boxes, float* __restrict__ partials)
{
    __shared__ int4  s_boxes[B_ * NBOX];
    __shared__ float sS[THREADS];
    __shared__ float sN[THREADS];

    const int tid = threadIdx.x;
    stage_boxes(gt_boxes, s_boxes, tid);

    const int qid = blockIdx.x * THREADS + tid;
    float tS = 0.0f, tN = 0.0f;

    if (qid < NQUAD) {
        int b, h, w0;
        decode_quad(qid, b, h, w0);
        const unsigned qmask = quad_mask(&s_boxes[b * NBOX], h, w0);
        if (qmask) {
            const size_t base = (size_t)b * ((size_t)C_ * HW_) + (size_t)h * W_ + w0;
            const float* xp = input + base;
            const float* tp = target + base;
            float s0 = 0.f, s1 = 0.f, s2 = 0.f, s3 = 0.f;
            unsigned nz = 0u;
#pragma unroll 8
            for (int c = 0; c < C_; ++c) {
                const v4f xv = __builtin_nontemporal_load((const v4f*)(xp + (size_t)c * HW_));
                const v4f tv = __builtin_nontemporal_load((const v4f*)(tp + (size_t)c * HW_));
                nz |= (tv.x != 0.f ? 1u : 0u) | (tv.y != 0.f ? 2u : 0u)
                    | (tv.z != 0.f ? 4u : 0u) | (tv.w != 0.f ? 8u : 0u);
                const float d0 = (tv.x == tv.x) ? (xv.x - tv.x) : 0.f;
                const float d1 = (tv.y == tv.y) ? (xv.y - tv.y) : 0.f;
                const float d2 = (tv.z == tv.z) ? (xv.z - tv.z) : 0.f;
                const float d3 = (tv.w == tv.w) ? (xv.w - tv.w) : 0.f;
                s0 = fmaf(d0, d0, s0);
                s1 = fmaf(d1, d1, s1);
                s2 = fmaf(d2, d2, s2);
                s3 = fmaf(d3, d3, s3);
            }
            const unsigned act = qmask & nz;
            if (act & 1u) { tS += s0; tN += 1.0f; }
            if (act & 2u) { tS += s1; tN += 1.0f; }
            if (act & 4u) { tS += s2; tN += 1.0f; }
            if (act & 8u) { tS += s3; tN += 1.0f; }
        }
    }

    sS[tid] = tS;
    sN[tid] = tN;
    __syncthreads();
    for (int s = THREADS / 2; s > 0; s >>= 1) {
        if (tid < s) {
            sS[tid] += sS[tid + s];
            sN[tid] += sN[tid + s];
        }
        __syncthreads();
    }
    if (tid == 0) {
        partials[2 * blockIdx.x + 0] = sS[0];
        partials[2 * blockIdx.x + 1] = sN[0];
    }
}

// ---------------------------------------------------------------------------
// Finalize: single block, fixed-order reduction of per-block partials.
//   out = 0.5 * S / (N * C * B)
// ---------------------------------------------------------------------------
extern "C" __global__ void __launch_bounds__(THREADS)
mfl_finalize(const float* __restrict__ partials, float* __restrict__ out)
{
    __shared__ float sS[THREADS];
    __shared__ float sN[THREADS];
    const int tid = threadIdx.x;

    float S = 0.0f, N = 0.0f;
    for (int i = tid; i < NBLK; i += THREADS) {   // fixed order per thread
        S += partials[2 * i + 0];
        N += partials[2 * i + 1];
    }
    sS[tid] = S;
    sN[tid] = N;
    __syncthreads();
    for (int s = THREADS / 2; s > 0; s >>= 1) {
        if (tid < s) {
            sS[tid] += sS[tid + s];
            sN[tid] += sN[tid + s];
        }
        __syncthreads();
    }
    if (tid == 0) {
        out[0] = 0.5f * sS[0] / (sN[0] * (float)(C_ * B_));
    }
}

// ---------------------------------------------------------------------------
extern "C" void kernel_launch(void* const* d_in, const int* in_sizes, int n_in,
                              void* d_out, int out_size, void* d_ws, size_t ws_size,
                              hipStream_t stream)
{
    (void)in_sizes; (void)n_in; (void)out_size;
    const float* input  = (const float*)d_in[0];
    const float* target = (const float*)d_in[1];
    const int*   boxes  = (const int*)d_in[2];
    float*       out    = (float*)d_out;
    char*        ws     = (char*)d_ws;

    if (ws_size >= WS_TOTAL) {
        v4f*      ssq  = (v4f*)(ws);
        unsigned* nz   = (unsigned*)(ws + WS_SSQ_BYTES);
        float*    part = (float*)(ws + WS_SSQ_BYTES + WS_NZ_BYTES);
        dim3 grid(NBLK, CC);
        mfl_stream<<<grid, THREADS, 0, stream>>>(input, target, boxes, ssq, nz);
        mfl_combine<<<NBLK, THREADS, 0, stream>>>(boxes, ssq, nz, part);
        mfl_finalize<<<1, THREADS, 0, stream>>>(part, out);
    } else {
        float* part = (float*)ws;                 // 2*NBLK floats
        mfl_fused<<<NBLK, THREADS, 0, stream>>>(input, target, boxes, part);
        mfl_finalize<<<1, THREADS, 0, stream>>>(part, out);
    }
}